// VenomSparseLinear_56813827392139
// MI455X (gfx1250) — compile-verified
//
#include <hip/hip_runtime.h>

typedef float v2f __attribute__((ext_vector_type(2)));
typedef float v4f __attribute__((ext_vector_type(4)));
typedef float v8f __attribute__((ext_vector_type(8)));

#define M_TOTAL 16384   // B*S = 8*2048
#define N_DIM   4096    // out_features
#define K_DIM   4096    // in_features (raw)
#define K_EFF   1024    // active columns: 2 of every 8
#define BLK_M   128
#define BLK_N   128
#define BLK_K   16      // gathered k per stage = 8 groups of 8 raw cols
#define KP      18      // padded LDS row stride (even -> 8B-aligned v2f, bank-spreading)

// y[m,n] = sum_j x[m, kmap(j)] * w[n, kmap(j)] + bias[n],
// kmap(j) = (j>>1)*8 + (j&1)*2   (static 2:8 VENOM pattern: positions 0 and 2)
__global__ __launch_bounds__(256) void venom_sparse_linear_f32wmma(
    const float* __restrict__ x, const float* __restrict__ w,
    const float* __restrict__ bias, float* __restrict__ out)
{
    // M-major gathered tiles, double buffered: fragment = 2 contiguous floats -> ds_load_b64
    __shared__ float As[2][BLK_M][KP];
    __shared__ float Bs[2][BLK_N][KP];

    const int tid  = threadIdx.x;
    const int lane = tid & 31;
    const int wave = tid >> 5;       // 8 waves
    const int wm   = wave & 3;       // 4 waves along M (32 rows each)
    const int wn   = wave >> 2;      // 2 waves along N (64 cols each)
    const int half = lane >> 4;      // 0 = lanes 0-15, 1 = lanes 16-31
    const int r    = lane & 15;

    const int blockN = blockIdx.x * BLK_N;
    const int blockM = blockIdx.y * BLK_M;

    // staging assignment: pair index p = tid + i*256; g = p&7 (k-group), row = p>>3
    const int g0   = tid & 7;
    const int row0 = tid >> 3;       // 0..31, +32 per i
    const int nk   = K_EFF / BLK_K;  // 64 stages

    v8f acc[2][4] = {};              // 2 (M) x 4 (N) tiles of 16x16 f32

    // ---- prologue: stage kt = 0 into buffer 0 ----
    #pragma unroll
    for (int i = 0; i < 4; ++i) {
        const int m = row0 + i * 32;
        const v4f xa = *(const v4f*)&x[(long)(blockM + m) * K_DIM + (long)g0 * 8];
        v2f t; t.x = xa.x; t.y = xa.z;               // keep positions 0 and 2
        *(v2f*)&As[0][m][g0 * 2] = t;
        const v4f wa = *(const v4f*)&w[(long)(blockN + m) * K_DIM + (long)g0 * 8];
        v2f u; u.x = wa.x; u.y = wa.z;
        *(v2f*)&Bs[0][m][g0 * 2] = u;
    }
    __syncthreads();

    for (int kt = 0; kt < nk; ++kt) {
        const int cur = kt & 1;
        const int nxt = cur ^ 1;

        // ---- stage kt+1 into the other buffer (overlaps with WMMA below) ----
        if (kt + 1 < nk) {
            #pragma unroll
            for (int i = 0; i < 4; ++i) {
                const int m = row0 + i * 32;
                const long xoff = (long)(blockM + m) * K_DIM + (long)((kt + 1) * 8 + g0) * 8;
                const v4f xa = *(const v4f*)&x[xoff];
                v2f t; t.x = xa.x; t.y = xa.z;
                *(v2f*)&As[nxt][m][g0 * 2] = t;
                const long woff = (long)(blockN + m) * K_DIM + (long)((kt + 1) * 8 + g0) * 8;
                const v4f wa = *(const v4f*)&w[woff];
                v2f u; u.x = wa.x; u.y = wa.z;
                *(v2f*)&Bs[nxt][m][g0 * 2] = u;
                if (i == 0 && kt + 2 < nk) {         // prefetch stage kt+2 lines
                    __builtin_prefetch(&x[xoff + 64], 0, 3);
                    __builtin_prefetch(&w[woff + 64], 0, 3);
                }
            }
        }

        // ---- 4 WMMA k-steps of K=4 on the current buffer ----
        #pragma unroll
        for (int ks = 0; ks < BLK_K; ks += 4) {
            const int krow = ks + half * 2;          // lanes 0-15: K=0/1; 16-31: K=2/3
            v2f a[2], b[4];
            #pragma unroll
            for (int mt = 0; mt < 2; ++mt)
                a[mt] = *(const v2f*)&As[cur][wm * 32 + mt * 16 + r][krow];
            #pragma unroll
            for (int nt = 0; nt < 4; ++nt)
                b[nt] = *(const v2f*)&Bs[cur][wn * 64 + nt * 16 + r][krow];
            #pragma unroll
            for (int mt = 0; mt < 2; ++mt)
                #pragma unroll
                for (int nt = 0; nt < 4; ++nt)
                    acc[mt][nt] = __builtin_amdgcn_wmma_f32_16x16x4_f32(
                        false, a[mt], false, b[nt],
                        (short)0, acc[mt][nt], false, false);
        }
        __syncthreads();
    }

    // ---- epilogue: fused bias add + store (C/D layout: VGPR v -> M = v + half*8) ----
    #pragma unroll
    for (int nt = 0; nt < 4; ++nt) {
        const int col = blockN + wn * 64 + nt * 16 + r;
        const float bv = bias[col];
        #pragma unroll
        for (int mt = 0; mt < 2; ++mt) {
            const int rowBase = blockM + wm * 32 + mt * 16 + half * 8;
            #pragma unroll
            for (int v = 0; v < 8; ++v)
                out[(long)(rowBase + v) * N_DIM + col] = acc[mt][nt][v] + bv;
        }
    }
}

extern "C" void kernel_launch(void* const* d_in, const int* in_sizes, int n_in,
                              void* d_out, int out_size, void* d_ws, size_t ws_size,
                              hipStream_t stream) {
    const float* x    = (const float*)d_in[0];   // [8,2048,4096] f32
    const float* w    = (const float*)d_in[1];   // [4096,4096]  f32
    const float* bias = (const float*)d_in[2];   // [4096]       f32
    float* out        = (float*)d_out;           // [8,2048,4096] f32

    dim3 grid(N_DIM / BLK_N, M_TOTAL / BLK_M);   // 32 x 128
    dim3 block(256);                             // 8 wave32s
    venom_sparse_linear_f32wmma<<<grid, block, 0, stream>>>(x, w, bias, out);
}